// CustomLoss_81793357185642
// MI455X (gfx1250) — compile-verified
//
#include <hip/hip_runtime.h>

typedef __attribute__((ext_vector_type(4))) float f4;
typedef __attribute__((ext_vector_type(2))) float v2f;
typedef __attribute__((ext_vector_type(8))) float v8f;

#define THREADS 256
#define BLOCKS  2048
#define NWAVES  (THREADS / 32)

// Kernel 1: streaming squared-diff accumulation.
// Each lane loads float4 of predicted/target (global_load_b128, non-temporal),
// squares the diffs with VALU, and folds them through v_wmma_f32_16x16x4_f32
// with B = ones into a chained f32 accumulator matrix. Sum is layout-invariant,
// so any assignment of loaded values to A-matrix slots is correct.
__global__ __launch_bounds__(THREADS) void sqdiff_partial_kernel(
    const f4* __restrict__ pred, const f4* __restrict__ targ,
    float* __restrict__ partial, int nvec) {
  const int tid    = blockIdx.x * THREADS + threadIdx.x;
  const int stride = BLOCKS * THREADS;
  const int iters  = nvec / stride;           // uniform across all waves

  v8f acc0 = {};
  v8f acc1 = {};
  const v2f ones = {1.0f, 1.0f};

  int idx = tid;
  for (int it = 0; it < iters; ++it, idx += stride) {
    f4 p = __builtin_nontemporal_load(&pred[idx]);
    f4 t = __builtin_nontemporal_load(&targ[idx]);
    float dx = p.x - t.x;
    float dy = p.y - t.y;
    float dz = p.z - t.z;
    float dw = p.w - t.w;
    v2f a0 = {dx * dx, dy * dy};
    v2f a1 = {dz * dz, dw * dw};
    // D = A x ones + C : every C slot accumulates a 4-element row sum.
    acc0 = __builtin_amdgcn_wmma_f32_16x16x4_f32(false, a0, false, ones,
                                                 (short)0, acc0, false, false);
    acc1 = __builtin_amdgcn_wmma_f32_16x16x4_f32(false, a1, false, ones,
                                                 (short)0, acc1, false, false);
  }

  // Tail (empty for the reference shape; plain FMA path, may diverge safely).
  float tacc = 0.0f;
  for (int i = iters * stride + tid; i < nvec; i += stride) {
    f4 p = __builtin_nontemporal_load(&pred[i]);
    f4 t = __builtin_nontemporal_load(&targ[i]);
    float dx = p.x - t.x;
    float dy = p.y - t.y;
    float dz = p.z - t.z;
    float dw = p.w - t.w;
    tacc = fmaf(dx, dx, tacc);
    tacc = fmaf(dy, dy, tacc);
    tacc = fmaf(dz, dz, tacc);
    tacc = fmaf(dw, dw, tacc);
  }
  // Wave-reduce the tail scalar (fixed order -> deterministic).
  #pragma unroll
  for (int off = 16; off > 0; off >>= 1) tacc += __shfl_xor(tacc, off, 32);

  // Fold the WMMA accumulators. After summing the 8 C VGPRs:
  //   lanes 0-15  hold sum over rows M=0..7   (replicated across columns)
  //   lanes 16-31 hold sum over rows M=8..15
  float v = 0.0f;
  #pragma unroll
  for (int k = 0; k < 8; ++k) v += acc0[k] + acc1[k];
  v += __shfl_xor(v, 16, 32);   // every lane now holds the wave's WMMA total

  float wave_total = v + tacc;

  __shared__ float wsum[NWAVES];
  const int wave = threadIdx.x >> 5;
  const int lane = threadIdx.x & 31;
  if (lane == 0) wsum[wave] = wave_total;
  __syncthreads();

  if (threadIdx.x == 0) {
    float s = 0.0f;
    #pragma unroll
    for (int w = 0; w < NWAVES; ++w) s += wsum[w];   // fixed order
    partial[blockIdx.x] = s;
  }
}

// Kernel 2: deterministic fixed-order reduction of the 2048 block partials.
__global__ __launch_bounds__(THREADS) void finalize_kernel(
    const float* __restrict__ partial, float* __restrict__ out) {
  __shared__ float s[THREADS];
  float v = 0.0f;
  for (int i = threadIdx.x; i < BLOCKS; i += THREADS) v += partial[i];
  s[threadIdx.x] = v;
  __syncthreads();
  #pragma unroll
  for (int off = THREADS / 2; off > 0; off >>= 1) {
    if (threadIdx.x < off) s[threadIdx.x] += s[threadIdx.x + off];
    __syncthreads();
  }
  if (threadIdx.x == 0) out[0] = s[0] * 0.5f;
}

extern "C" void kernel_launch(void* const* d_in, const int* in_sizes, int n_in,
                              void* d_out, int out_size, void* d_ws, size_t ws_size,
                              hipStream_t stream) {
  const f4* pred = (const f4*)d_in[0];
  const f4* targ = (const f4*)d_in[1];
  float* out     = (float*)d_out;
  float* partial = (float*)d_ws;          // needs BLOCKS * 4 = 8 KB of scratch

  const int nvec = in_sizes[0] / 4;       // float4 count = 33,554,432

  sqdiff_partial_kernel<<<BLOCKS, THREADS, 0, stream>>>(pred, targ, partial, nvec);
  finalize_kernel<<<1, THREADS, 0, stream>>>(partial, out);
}